// ObjectDetection_57621281243681
// MI455X (gfx1250) — compile-verified
//
#include <hip/hip_runtime.h>
#include <hip/hip_bf16.h>
#include <stdint.h>

// ---------------------------------------------------------------------------
// Mask-RCNN post-processing for MI455X (gfx1250, wave32).
//   N=6000 proposals, C=81 classes, masks 14x14x81 per proposal, 300 outputs.
// Latency-bound (serial NMS) + small-gather bandwidth; no GEMM -> no WMMA.
// Uses CDNA5 async-to-LDS data movers + 320KB LDS + wave32 shuffles.
// ---------------------------------------------------------------------------

namespace {
constexpr int   kN       = 6000;
constexpr int   kC       = 81;
constexpr int   kMaskRow = 14 * 14 * 81;   // 15876 floats (= 16B * 3969)
constexpr int   kPad     = 300;
constexpr float kThresh  = 0.5f;
}

static __device__ __forceinline__ float neg_inf() { return -__builtin_inff(); }

// Low 32 bits of a flat pointer into __shared__ == LDS byte offset
// (a flat LDS address carries the LDS offset in its low 32 bits; see ISA 10.2).
static __device__ __forceinline__ unsigned lds_byte_off(const void* p) {
  return (unsigned)(uintptr_t)p;
}

// CDNA5 async global->LDS movers (GV mode: vdst = LDS offset VGPR,
// vaddr = 64-bit global address, saddr = off). Tracked by ASYNCcnt.
static __device__ __forceinline__ void async_ld_lds_b128(unsigned loff, const void* g) {
  uint64_t ga = (uint64_t)(uintptr_t)g;
  asm volatile("global_load_async_to_lds_b128 %0, %1, off"
               :: "v"(loff), "v"(ga) : "memory");
}
static __device__ __forceinline__ void async_ld_lds_b32(unsigned loff, const void* g) {
  uint64_t ga = (uint64_t)(uintptr_t)g;
  asm volatile("global_load_async_to_lds_b32 %0, %1, off"
               :: "v"(loff), "v"(ga) : "memory");
}
static __device__ __forceinline__ void wait_async0() {
  asm volatile("s_wait_asynccnt 0" ::: "memory");
}

// ---------------------------------------------------------------------------
// Kernel 1: per-proposal decode. One wave32 per proposal.
//   - argmax over 81 class scores (first-occurrence tie-break == jnp.argmax)
//   - decode + clip the winning class box only
//   - max score over classes 1..80
// ---------------------------------------------------------------------------
__global__ __launch_bounds__(256) void k_decode(
    const float* __restrict__ meta, const float* __restrict__ deltas,
    const float* __restrict__ proposals, const float* __restrict__ scores,
    float* __restrict__ pbox, float* __restrict__ maxsc) {
  int gt   = blockIdx.x * blockDim.x + threadIdx.x;
  int row  = gt >> 5;
  int lane = gt & 31;
  if (row >= kN) return;

  const float* srow = scores + (size_t)row * kC;
  float bestv = neg_inf(); int besti = 0;
  float mv = neg_inf();
  for (int c = lane; c < kC; c += 32) {       // ascending c: strict '>' keeps first max
    float v = srow[c];
    if (v > bestv) { bestv = v; besti = c; }
    if (c >= 1) mv = fmaxf(mv, v);
  }
  for (int off = 16; off > 0; off >>= 1) {    // wave32 reduce
    float ov = __shfl_xor(bestv, off, 32);
    int   oi = __shfl_xor(besti, off, 32);
    if (ov > bestv || (ov == bestv && oi < besti)) { bestv = ov; besti = oi; }
    mv = fmaxf(mv, __shfl_xor(mv, off, 32));
  }

  if (lane == 0) {
    float H = meta[0], W = meta[1], scale = meta[2];
    const float* p = proposals + (size_t)row * 4;
    float x1 = p[0] / scale, y1 = p[1] / scale;
    float x2 = p[2] / scale, y2 = p[3] / scale;
    float w  = x2 - x1 + 1.0f, h  = y2 - y1 + 1.0f;
    float cx = x1 + 0.5f * w,  cy = y1 + 0.5f * h;
    const float* d = deltas + (size_t)row * (4 * kC) + 4 * besti;
    float pcx = d[0] * w + cx;
    float pcy = d[1] * h + cy;
    float pw  = expf(d[2]) * w;
    float ph  = expf(d[3]) * h;
    float bx1 = pcx - 0.5f * pw, by1 = pcy - 0.5f * ph;
    float bx2 = pcx + 0.5f * pw, by2 = pcy + 0.5f * ph;
    bx1 = fminf(fmaxf(bx1, 0.0f), W - 1.0f);
    by1 = fminf(fmaxf(by1, 0.0f), H - 1.0f);
    bx2 = fminf(fmaxf(bx2, 0.0f), W - 1.0f);
    by2 = fminf(fmaxf(by2, 0.0f), H - 1.0f);
    *(float4*)(pbox + (size_t)row * 4) = make_float4(bx1, by1, bx2, by2);
    maxsc[row] = mv;
  }
}

// ---------------------------------------------------------------------------
// Kernel 2: greedy NMS, single 1024-thread workgroup (32 waves).
// Entire working set (6000 boxes AoS + 6000 scores = 120 KB) lives in LDS
// (fits only because CDNA5 gives 320 KB/WGP); staged in via async movers.
// ---------------------------------------------------------------------------
__global__ __launch_bounds__(1024) void k_nms(
    const float* __restrict__ pbox, const float* __restrict__ maxsc,
    int* __restrict__ idx_out, int* __restrict__ valid_out) {
  extern __shared__ char smem[];
  float* bx = (float*)smem;                   // 6000 * float4 (AoS boxes) = 96000 B
  float* s  = (float*)(smem + kN * 16);       // 6000 * float               = 24000 B
  __shared__ float rv[32];
  __shared__ int   ri[32];
  __shared__ float bb[5];                     // x1,y1,x2,y2,area of selected box
  __shared__ int   bsel;

  const int tid = threadIdx.x;

  // Async stage boxes + scores into LDS (ASYNCcnt-tracked).
  for (int n = tid; n < kN; n += 1024) {
    async_ld_lds_b128(lds_byte_off(bx + n * 4), pbox + (size_t)n * 4);
    async_ld_lds_b32 (lds_byte_off(s + n),      maxsc + n);
  }
  wait_async0();
  __syncthreads();

  for (int it = 0; it < kPad; ++it) {
    // ---- global argmax over s[0..N) : value desc, index asc on ties ----
    float bv = s[tid]; int bi = tid;          // tid < 1024 <= N always valid
    for (int n = tid + 1024; n < kN; n += 1024) {
      float v = s[n];
      if (v > bv) { bv = v; bi = n; }
    }
    for (int off = 16; off > 0; off >>= 1) {
      float ov = __shfl_xor(bv, off, 32);
      int   oi = __shfl_xor(bi, off, 32);
      if (ov > bv || (ov == bv && oi < bi)) { bv = ov; bi = oi; }
    }
    if ((tid & 31) == 0) { rv[tid >> 5] = bv; ri[tid >> 5] = bi; }
    __syncthreads();
    if (tid < 32) {
      bv = rv[tid]; bi = ri[tid];
      for (int off = 16; off > 0; off >>= 1) {
        float ov = __shfl_xor(bv, off, 32);
        int   oi = __shfl_xor(bi, off, 32);
        if (ov > bv || (ov == bv && oi < bi)) { bv = ov; bi = oi; }
      }
      if (tid == 0) {
        bsel = bi;
        float4 b = *(const float4*)(bx + bi * 4);
        bb[0] = b.x; bb[1] = b.y; bb[2] = b.z; bb[3] = b.w;
        bb[4] = (b.z - b.x + 1.0f) * (b.w - b.y + 1.0f);
        idx_out[it]   = bi;
        valid_out[it] = (bv > neg_inf()) ? 1 : 0;
      }
    }
    __syncthreads();

    // ---- suppress: iou > thresh or n == i  -> s[n] = -inf ----
    const int i = bsel;
    const float ax1 = bb[0], ay1 = bb[1], ax2 = bb[2], ay2 = bb[3], aarea = bb[4];
    for (int n = tid; n < kN; n += 1024) {
      float4 b = *(const float4*)(bx + n * 4);
      float xx1 = fmaxf(ax1, b.x), yy1 = fmaxf(ay1, b.y);
      float xx2 = fminf(ax2, b.z), yy2 = fminf(ay2, b.w);
      float inter = fmaxf(xx2 - xx1 + 1.0f, 0.0f) * fmaxf(yy2 - yy1 + 1.0f, 0.0f);
      float area  = (b.z - b.x + 1.0f) * (b.w - b.y + 1.0f);
      float iou   = inter / (aarea + area - inter);
      if (iou > kThresh || n == i) s[n] = neg_inf();
    }
    __syncthreads();
  }
}

// ---------------------------------------------------------------------------
// Kernel 3: gather outputs. One block per output slot k (300 blocks).
// Mask row (62 KB) async-staged into LDS, then streamed out as float4.
// ---------------------------------------------------------------------------
__global__ __launch_bounds__(256) void k_gather(
    const float* __restrict__ pbox, const float* __restrict__ scores,
    const float* __restrict__ masks, const int* __restrict__ idx,
    const int* __restrict__ valid, float* __restrict__ out) {
  extern __shared__ char smem[];
  float* lbuf = (float*)smem;                 // kMaskRow floats = 63504 B

  const int k   = blockIdx.x;
  const int tid = threadIdx.x;
  const int i   = idx[k];
  const int v   = valid[k];
  const float vf = v ? 1.0f : 0.0f;

  float* out_boxes  = out;
  float* out_scores = out + (size_t)kPad * 4;
  float* out_masks  = out + (size_t)kPad * 4 + (size_t)kPad * kC;

  if (tid < 4)  out_boxes [k * 4 + tid]  = pbox  [(size_t)i * 4 + tid]  * vf;
  if (tid < kC) out_scores[k * kC + tid] = scores[(size_t)i * kC + tid] * vf;

  float* dst = out_masks + (size_t)k * kMaskRow;
  if (v) {                                    // vf == 1.0: exact copy
    const float* src = masks + (size_t)i * kMaskRow;
    for (int t = tid; t < kMaskRow / 4; t += 256)
      async_ld_lds_b128(lds_byte_off(lbuf + t * 4), src + (size_t)t * 4);
    wait_async0();
    __syncthreads();
    for (int t = tid; t < kMaskRow / 4; t += 256)
      *(float4*)(dst + (size_t)t * 4) = *(const float4*)(lbuf + t * 4);
  } else {                                    // vf == 0: write zeros
    float4 z = make_float4(0.0f, 0.0f, 0.0f, 0.0f);
    for (int t = tid; t < kMaskRow / 4; t += 256)
      *(float4*)(dst + (size_t)t * 4) = z;
  }
}

// ---------------------------------------------------------------------------
extern "C" void kernel_launch(void* const* d_in, const int* in_sizes, int n_in,
                              void* d_out, int out_size, void* d_ws, size_t ws_size,
                              hipStream_t stream) {
  const float* meta      = (const float*)d_in[0];   // [1,3]   H,W,SCALE
  const float* deltas    = (const float*)d_in[1];   // [1,N,4C]
  const float* proposals = (const float*)d_in[2];   // [1,N,4]
  const float* scores    = (const float*)d_in[3];   // [1,N,C]
  const float* masks     = (const float*)d_in[4];   // [1,N,14,14,81]

  char* ws = (char*)d_ws;
  float* pbox   = (float*)(ws);                     //  96000 B
  float* maxsc  = (float*)(ws + 96000);             //  24000 B
  int*   idx    = (int*)  (ws + 120000);            //   1200 B
  int*   valid  = (int*)  (ws + 121216);            //   1200 B
  float* out    = (float*)d_out;

  // K1: 6000 waves, 8 waves (256 thr) per block.
  k_decode<<<(kN * 32 + 255) / 256, 256, 0, stream>>>(
      meta, deltas, proposals, scores, pbox, maxsc);

  // K2: single workgroup, 120 KB dynamic LDS (CDNA5 320KB/WGP).
  k_nms<<<1, 1024, kN * 20, stream>>>(pbox, maxsc, idx, valid);

  // K3: 300 blocks, 62 KB dynamic LDS mask staging buffer.
  k_gather<<<kPad, 256, kMaskRow * 4, stream>>>(
      pbox, scores, masks, idx, valid, out);
}